// Embedder_56676388438608
// MI455X (gfx1250) — compile-verified
//
#include <hip/hip_runtime.h>
#include <hip/hip_bf16.h>

// ---------------------------------------------------------------------------
// RGCN (basis-decomposed RelGraphConv x4 + BN, eval mode) for MI455X / gfx1250
//  - per-relation GEMMs via V_WMMA_F32_16X16X4_F32 (fp32-exact vs reference)
//  - relation-chunked Y buffer so gather/scatter stays L2-resident
//  - W staged into LDS once per workgroup via the Tensor Data Mover
//    (tensor_load_to_lds + s_wait_tensorcnt)
//  - W stored k-pair-interleaved ([K/2][64][2]) so each WMMA B fragment is a
//    single aligned ds_load_b64 (no register repacking in the inner loop)
// ---------------------------------------------------------------------------

typedef __attribute__((ext_vector_type(2))) float        v2f;
typedef __attribute__((ext_vector_type(8))) float        v8f;
typedef __attribute__((ext_vector_type(4))) unsigned int v4u;
typedef __attribute__((ext_vector_type(8))) int          v8i;
typedef __attribute__((ext_vector_type(4))) int          v4i;

#define N_NODES   50000
#define N_EDGES   800000
#define IN_DIM    16
#define H_DIM     64
#define N_REL     20
#define N_BASES   20
#define N_LAYERS  4
#define RCHUNK    4                      // relations per Y chunk (51.2 MB, L2-resident)
#define ROW_TILES (N_NODES / 16)         // 3125 (N divides 16 exactly)
#define WSTRIDE   (H_DIM * H_DIM)        // per-relation weight stride (max K = 64)
#define BN_EPS    1e-5f

// interleaved index for W element (k, o):  [k/2][o][k&1]
#define WIDX(k, o) ((((k) >> 1) * (H_DIM * 2)) + ((o) << 1) + ((k) & 1))

// ---------------------------------------------------------------------------
// LDS byte-offset of a __shared__ object (generic -> AS(3) addrspacecast).
// ---------------------------------------------------------------------------
__device__ __forceinline__ unsigned lds_off_u32(const void* p)
{
    return (unsigned)(unsigned long long)
        (__attribute__((address_space(3))) const void*)p;
}

// ---------------------------------------------------------------------------
// TDM: DMA nelems f32 (contiguous) from global to LDS. D# per ISA ch.8:
//  group0: count=1 | lds_addr | global_addr[56:0] | type=2
//  group1: data_size=4B, tensor_dim0=nelems, tensor_dim1=1,
//          tile_dim0=nelems, tile_dim1=1, tensor_dim0_stride=nelems
// 6-arg toolchain form: (g0, g1, g2, g3, g_extra, cpol).
// Issued once (caller guards to one wave); tracked by TENSORcnt.
// ---------------------------------------------------------------------------
__device__ __forceinline__ void tdm_load_f32(unsigned lds_byte_off,
                                             const float* gsrc, int nelems)
{
    unsigned long long ga = (unsigned long long)(uintptr_t)gsrc;
    v4u g0;
    g0.x = 1u;                                            // count=1 (user D#)
    g0.y = lds_byte_off;                                  // lds_addr
    g0.z = (unsigned)ga;                                  // global_addr[31:0]
    g0.w = (unsigned)((ga >> 32) & 0x01FFFFFFu)           // global_addr[56:32]
         | (2u << 30);                                    // type = 2 ("image")
    v8i g1;
    g1[0] = (int)(2u << 16);                              // data_size = 2 -> 4 bytes
    g1[1] = (int)((unsigned)(nelems & 0xFFFF) << 16);     // tensor_dim0[15:0]
    g1[2] = (int)(((unsigned)nelems >> 16) & 0xFFFFu)     // tensor_dim0[31:16]
          | (1 << 16);                                    // tensor_dim1 = 1
    g1[3] = (int)((unsigned)(nelems & 0xFFFF) << 16);     // tile_dim0
    g1[4] = 1;                                            // tile_dim1 = 1
    g1[5] = nelems;                                       // tensor_dim0_stride lo
    g1[6] = 0;
    g1[7] = 0;
    v4i z4 = {0, 0, 0, 0};
    v8i z8 = {0, 0, 0, 0, 0, 0, 0, 0};
    __builtin_amdgcn_tensor_load_to_lds(g0, g1, z4, z4, z8, 0);
}

// ---------------------------------------------------------------------------
// W_all[r] (interleaved) = sum_b comb[r,b] * V[b][k][o]
// ---------------------------------------------------------------------------
__global__ void compute_w_kernel(const float* __restrict__ comb,   // [R,B] (layer slice)
                                 const float* __restrict__ V,      // [B,K,64]
                                 float* __restrict__ Wall,         // [R][WSTRIDE] interleaved
                                 int K)
{
    const int r = blockIdx.x;
    const int total = K * H_DIM;
    for (int idx = threadIdx.x; idx < total; idx += blockDim.x) {
        float s = 0.f;
        #pragma unroll 4
        for (int b = 0; b < N_BASES; ++b)
            s += comb[r * N_BASES + b] * V[(size_t)b * total + idx];
        const int k = idx / H_DIM;
        const int o = idx % H_DIM;
        Wall[(size_t)r * WSTRIDE + WIDX(k, o)] = s;
    }
}

// Interleave a [K][64] weight matrix into the WIDX layout (for Wloop).
__global__ void interleave_w_kernel(const float* __restrict__ Wsrc, // [K,64]
                                    float* __restrict__ Wil,        // [WSTRIDE]
                                    int K)
{
    const int total = K * H_DIM;
    for (int idx = blockIdx.x * blockDim.x + threadIdx.x; idx < total;
         idx += gridDim.x * blockDim.x) {
        const int k = idx / H_DIM;
        const int o = idx % H_DIM;
        Wil[WIDX(k, o)] = Wsrc[idx];
    }
}

__global__ void zero_kernel(float* __restrict__ p, int n)
{
    int i = blockIdx.x * blockDim.x + threadIdx.x;
    if (i < n) p[i] = 0.f;
}

// ---------------------------------------------------------------------------
// Y[rc][n][:] = h[n][:] @ W_{r0+rc}  — one wave per 16x64 tile, fp32 WMMA.
// W_r (interleaved) staged to LDS by the TDM once per workgroup.
// A frag (16x4 f32): lane m = lane&15 holds row m; half-wave picks K pair.
// B/C/D: row striped across lanes (ISA 7.12.2); B fragment = one ds_load_b64.
// ---------------------------------------------------------------------------
__global__ void __launch_bounds__(128)
gemm_y_kernel(const float* __restrict__ h,     // [N,K]
              const float* __restrict__ Wall,  // [R][WSTRIDE] interleaved
              float* __restrict__ Y,           // [RCHUNK][N][64]
              int K, int r0)
{
    __shared__ float sW[H_DIM * H_DIM];        // 16 KB (K*64 floats used)

    const int lane = threadIdx.x & 31;
    const int wave = threadIdx.x >> 5;
    const int tile = blockIdx.x * 4 + wave;    // uniform per wave
    const int rc   = blockIdx.y;
    const float* __restrict__ W = Wall + (size_t)(r0 + rc) * WSTRIDE;

    if (threadIdx.x < 32) {                    // wave 0 drives the TDM
        tdm_load_f32(lds_off_u32(sW), W, K * H_DIM);
        __builtin_amdgcn_s_wait_tensorcnt(0);
    }
    __syncthreads();
    if (tile >= ROW_TILES) return;             // wave-uniform; EXEC stays all-1s

    const int m    = lane & 15;
    const int kh   = (lane >> 4) << 1;         // 0 or 2
    const int row0 = tile * 16;

    v8f acc0 = {}; v8f acc1 = {}; v8f acc2 = {}; v8f acc3 = {};
    for (int k0 = 0; k0 < K; k0 += 4) {
        const v2f a = *(const v2f*)(h + (size_t)(row0 + m) * K + (k0 + kh));
        // interleaved: pair index (k0+kh)/2, column n = c*16+m at (n<<1)
        const float* bp = sW + (((k0 + kh) >> 1) * (H_DIM * 2)) + (m << 1);
        const v2f b0 = *(const v2f*)(bp + 0);
        const v2f b1 = *(const v2f*)(bp + 32);
        const v2f b2 = *(const v2f*)(bp + 64);
        const v2f b3 = *(const v2f*)(bp + 96);
        acc0 = __builtin_amdgcn_wmma_f32_16x16x4_f32(false, a, false, b0, (short)0, acc0, false, false);
        acc1 = __builtin_amdgcn_wmma_f32_16x16x4_f32(false, a, false, b1, (short)0, acc1, false, false);
        acc2 = __builtin_amdgcn_wmma_f32_16x16x4_f32(false, a, false, b2, (short)0, acc2, false, false);
        acc3 = __builtin_amdgcn_wmma_f32_16x16x4_f32(false, a, false, b3, (short)0, acc3, false, false);
    }

    float* yb = Y + ((size_t)rc * N_NODES + row0) * H_DIM;
    const int rofs = (lane >> 4) << 3;         // 0 or 8
    #pragma unroll
    for (int j = 0; j < 8; ++j) {
        float* yr = yb + (size_t)(rofs + j) * H_DIM + m;
        yr[0]  = acc0[j];
        yr[16] = acc1[j];
        yr[32] = acc2[j];
        yr[48] = acc3[j];
    }
}

// ---------------------------------------------------------------------------
// agg[dst] += Y[etype - r0][src]  for edges whose etype is in this chunk.
// 16 threads per edge, float4 gather (Y chunk L2-resident), 4 f32 atomics.
// ---------------------------------------------------------------------------
__global__ void __launch_bounds__(256)
edge_scatter_kernel(const int* __restrict__ src,
                    const int* __restrict__ dst,
                    const int* __restrict__ etype,
                    const float* __restrict__ Y,   // [RCHUNK][N][64]
                    float* __restrict__ agg,       // [N][64]
                    int r0)
{
    const int tid = blockIdx.x * blockDim.x + threadIdx.x;
    const int e = tid >> 4;
    if (e >= N_EDGES) return;
    const int r = etype[e] - r0;
    if ((unsigned)r >= (unsigned)RCHUNK) return;
    const int q = (tid & 15) << 2;
    const float4 v = *(const float4*)(Y + ((size_t)r * N_NODES + src[e]) * H_DIM + q);
    float* a = agg + (size_t)dst[e] * H_DIM + q;
    atomicAdd(a + 0, v.x);
    atomicAdd(a + 1, v.y);
    atomicAdd(a + 2, v.z);
    atomicAdd(a + 3, v.w);
}

// ---------------------------------------------------------------------------
// out = BN( agg + h @ Wloop + bias ) [, ReLU]  — WMMA self-loop GEMM fused
// with the elementwise epilogue. One wave per 16x64 tile; Wloop (interleaved)
// staged via TDM.
// ---------------------------------------------------------------------------
__global__ void __launch_bounds__(128)
selfloop_bn_kernel(const float* __restrict__ h,      // [N,K]
                   const float* __restrict__ Wil,    // [WSTRIDE] interleaved Wloop
                   const float* __restrict__ agg,    // [N,64]
                   const float* __restrict__ bias,   // [64]
                   const float* __restrict__ gamma_, // [64]
                   const float* __restrict__ beta_,  // [64]
                   const float* __restrict__ mean_,  // [64]
                   const float* __restrict__ var_,   // [64]
                   float* __restrict__ out,          // [N,64]
                   int K, int do_relu)
{
    __shared__ float sW[H_DIM * H_DIM];

    const int lane = threadIdx.x & 31;
    const int wave = threadIdx.x >> 5;
    const int tile = blockIdx.x * 4 + wave;

    if (threadIdx.x < 32) {
        tdm_load_f32(lds_off_u32(sW), Wil, K * H_DIM);
        __builtin_amdgcn_s_wait_tensorcnt(0);
    }
    __syncthreads();
    if (tile >= ROW_TILES) return;

    const int m    = lane & 15;
    const int kh   = (lane >> 4) << 1;
    const int row0 = tile * 16;

    v8f acc0 = {}; v8f acc1 = {}; v8f acc2 = {}; v8f acc3 = {};
    for (int k0 = 0; k0 < K; k0 += 4) {
        const v2f a = *(const v2f*)(h + (size_t)(row0 + m) * K + (k0 + kh));
        const float* bp = sW + (((k0 + kh) >> 1) * (H_DIM * 2)) + (m << 1);
        const v2f b0 = *(const v2f*)(bp + 0);
        const v2f b1 = *(const v2f*)(bp + 32);
        const v2f b2 = *(const v2f*)(bp + 64);
        const v2f b3 = *(const v2f*)(bp + 96);
        acc0 = __builtin_amdgcn_wmma_f32_16x16x4_f32(false, a, false, b0, (short)0, acc0, false, false);
        acc1 = __builtin_amdgcn_wmma_f32_16x16x4_f32(false, a, false, b1, (short)0, acc1, false, false);
        acc2 = __builtin_amdgcn_wmma_f32_16x16x4_f32(false, a, false, b2, (short)0, acc2, false, false);
        acc3 = __builtin_amdgcn_wmma_f32_16x16x4_f32(false, a, false, b3, (short)0, acc3, false, false);
    }

    const int rofs = (lane >> 4) << 3;   // 0 or 8
    #pragma unroll
    for (int c = 0; c < 4; ++c) {
        const int col = c * 16 + m;
        const float bsc = gamma_[col] * rsqrtf(var_[col] + BN_EPS);
        const float bof = beta_[col] - mean_[col] * bsc;
        const float bi  = bias[col];
        const v8f acc = (c == 0) ? acc0 : (c == 1) ? acc1 : (c == 2) ? acc2 : acc3;
        #pragma unroll
        for (int j = 0; j < 8; ++j) {
            const int row = row0 + rofs + j;
            float v = agg[(size_t)row * H_DIM + col] + acc[j] + bi;
            v = v * bsc + bof;
            if (do_relu) v = fmaxf(v, 0.f);
            out[(size_t)row * H_DIM + col] = v;
        }
    }
}

// ---------------------------------------------------------------------------
extern "C" void kernel_launch(void* const* d_in, const int* in_sizes, int n_in,
                              void* d_out, int out_size, void* d_ws, size_t ws_size,
                              hipStream_t stream)
{
    const float* x      = (const float*)d_in[0];   // [N,16]
    const int*   src    = (const int*)  d_in[1];   // [E]
    const int*   dst    = (const int*)  d_in[2];   // [E]
    const int*   etype  = (const int*)  d_in[3];   // [E]
    const float* V0     = (const float*)d_in[4];   // [B,16,64]
    const float* Vh     = (const float*)d_in[5];   // [L-1,B,64,64]
    const float* comb   = (const float*)d_in[6];   // [L,R,B]
    const float* Wloop0 = (const float*)d_in[7];   // [16,64]
    const float* Wlooph = (const float*)d_in[8];   // [L-1,64,64]
    const float* bias   = (const float*)d_in[9];   // [L,64]
    const float* bng    = (const float*)d_in[10];  // [L,64]
    const float* bnb    = (const float*)d_in[11];  // [L,64]
    const float* bnm    = (const float*)d_in[12];  // [L,64]
    const float* bnv    = (const float*)d_in[13];  // [L,64]
    float* out = (float*)d_out;

    const size_t NH = (size_t)N_NODES * H_DIM;     // 3.2M floats
    float* ws    = (float*)d_ws;
    float* hbuf0 = ws;                              // [N,64]
    float* hbuf1 = ws + NH;                         // [N,64]
    float* agg   = ws + 2 * NH;                     // [N,64]
    float* Wall  = ws + 3 * NH;                     // [R][WSTRIDE] interleaved
    float* Wlil  = Wall + (size_t)N_REL * WSTRIDE;  // [WSTRIDE] interleaved Wloop
    float* Y     = Wlil + WSTRIDE;                  // [RCHUNK][N][64]
    float* hping[2] = { hbuf0, hbuf1 };

    const int gemm_blocks = (ROW_TILES + 3) / 4;                 // 782
    const int edge_blocks = (N_EDGES * 16 + 255) / 256;          // 50000
    const int zero_blocks = ((int)NH + 255) / 256;               // 12500

    for (int layer = 0; layer < N_LAYERS; ++layer) {
        const int K = (layer == 0) ? IN_DIM : H_DIM;
        const float* hin = (layer == 0) ? x : hping[(layer - 1) & 1];
        float* hout = (layer == N_LAYERS - 1) ? out : hping[layer & 1];
        const float* V  = (layer == 0) ? V0 : Vh + (size_t)(layer - 1) * N_BASES * H_DIM * H_DIM;
        const float* Wl = (layer == 0) ? Wloop0 : Wlooph + (size_t)(layer - 1) * H_DIM * H_DIM;
        const float* cb = comb + (size_t)layer * N_REL * N_BASES;

        compute_w_kernel<<<N_REL, 256, 0, stream>>>(cb, V, Wall, K);
        interleave_w_kernel<<<16, 256, 0, stream>>>(Wl, Wlil, K);
        zero_kernel<<<zero_blocks, 256, 0, stream>>>(agg, (int)NH);

        for (int r0 = 0; r0 < N_REL; r0 += RCHUNK) {
            dim3 grid(gemm_blocks, RCHUNK);
            gemm_y_kernel<<<grid, 128, 0, stream>>>(hin, Wall, Y, K, r0);
            edge_scatter_kernel<<<edge_blocks, 256, 0, stream>>>(src, dst, etype, Y, agg, r0);
        }

        selfloop_bn_kernel<<<gemm_blocks, 128, 0, stream>>>(
            hin, Wlil, agg,
            bias + (size_t)layer * H_DIM,
            bng  + (size_t)layer * H_DIM,
            bnb  + (size_t)layer * H_DIM,
            bnm  + (size_t)layer * H_DIM,
            bnv  + (size_t)layer * H_DIM,
            hout, K, (layer == N_LAYERS - 1) ? 1 : 0);
    }
}